// LGE_BilinearLM_45337674777285
// MI455X (gfx1250) — compile-verified
//
#include <hip/hip_runtime.h>

typedef __attribute__((ext_vector_type(2))) float v2f;
typedef __attribute__((ext_vector_type(8))) float v8f;

#define X_SITES (16*16*16*32)   // 131072 lattice sites

// out[i,x,n,m] = sum_{j,k,r} w[j,k,i] * f1[j,x,n,r] * f2[k,x,r,m]   (complex64)
//
// Per site (one wave32 each):
//   Stage 1 (VALU fp32): P[jk,nm] = (f1[j] @ f2[k])[nm], jk = j*4+k.
//     Each lane owns column nm = lane%16 (clamped to <9) and the 8 jk pairs
//     its half-wave feeds into the WMMA B-fragments.
//   Stage 2 (matrix pipe): real-embedded complex GEMM
//     D(16x16,f32) = A(16x32, weights, constant) x B(32x16, P per site)
//     as a chain of 8x V_WMMA_F32_16X16X4_F32 (full fp32 precision).
__global__ __launch_bounds__(256) void lge_bilinear_wmma_kernel(
    const float* __restrict__ f1,   // [4][X][3][3] complex64 -> float pairs
    const float* __restrict__ f2,   // [4][X][3][3] complex64
    const float* __restrict__ w,    // [4][4][8]    complex64
    float* __restrict__ out)        // [8][X][3][3] complex64
{
  const int lane  = threadIdx.x & 31;
  const int half  = lane >> 4;          // 0: lanes 0-15, 1: lanes 16-31
  const int M     = lane & 15;          // A-row / B-col / D-col index
  const int wave  = blockIdx.x * (blockDim.x >> 5) + (threadIdx.x >> 5);
  const int nwave = gridDim.x * (blockDim.x >> 5);

  // ---- A fragments (weights; identical for every site, hoisted) ----------
  // A[m,kk]: m<8 -> row of Re(out[i=m]), m>=8 -> row of Im(out[i=m-8]).
  //          kk<16 -> multiplies P_re[jk=kk], kk>=16 -> multiplies P_im[jk-16].
  //   top=0: [ wr , -wi ]   top=1: [ wi , wr ]
  // fp32 16x16x4 A layout: lane holds M=lane%16, v0/v1 = K = (4s+2*half)+{0,1}.
  v2f afrag[8];
  {
    const int i   = M & 7;
    const int top = M >> 3;
#pragma unroll
    for (int s = 0; s < 8; ++s) {
      float av[2];
#pragma unroll
      for (int e = 0; e < 2; ++e) {
        const int kk   = 4 * s + 2 * half + e;
        const int jk   = kk & 15;
        const int cplx = kk >> 4;
        const float wr = w[(jk * 8 + i) * 2 + 0];
        const float wi = w[(jk * 8 + i) * 2 + 1];
        av[e] = (top == 0) ? ((cplx == 0) ? wr : -wi)
                           : ((cplx == 0) ? wi :  wr);
      }
      afrag[s] = (v2f){av[0], av[1]};
    }
  }

  // Column owned by this lane (cols 9..15 compute junk into unused D cols;
  // clamp indices so their loads stay in bounds).
  const int nm = (M < 9) ? M : 8;
  const int n  = nm / 3;
  const int m  = nm % 3;

  for (int x = wave; x < X_SITES; x += nwave) {
    // f1[j, x, n, 0..2] for all j (row n of each f1 matrix)
    float ar[4][3], ai[4][3];
#pragma unroll
    for (int j = 0; j < 4; ++j) {
      const float* p = f1 + (((size_t)j * X_SITES + x) * 9 + n * 3) * 2;
#pragma unroll
      for (int r = 0; r < 3; ++r) { ar[j][r] = p[2 * r]; ai[j][r] = p[2 * r + 1]; }
    }
    // f2[k, x, 0..2, m] for this half-wave's two k values (k = 2*half + e)
    float br[2][3], bi[2][3];
#pragma unroll
    for (int e = 0; e < 2; ++e) {
      const int k = 2 * half + e;
      const float* p = f2 + (((size_t)k * X_SITES + x) * 9 + m) * 2;
#pragma unroll
      for (int r = 0; r < 3; ++r) { br[e][r] = p[6 * r]; bi[e][r] = p[6 * r + 1]; }
    }

    // Stage 1: P[jk] for jk = 4*q + 2*half + e  (j = q, k = 2*half + e)
    float pr[8], pi[8];
#pragma unroll
    for (int q = 0; q < 4; ++q) {
#pragma unroll
      for (int e = 0; e < 2; ++e) {
        float sr = 0.f, si = 0.f;
#pragma unroll
        for (int r = 0; r < 3; ++r) {
          sr = __builtin_fmaf(ar[q][r], br[e][r], sr);
          sr = __builtin_fmaf(-ai[q][r], bi[e][r], sr);
          si = __builtin_fmaf(ar[q][r], bi[e][r], si);
          si = __builtin_fmaf(ai[q][r], br[e][r], si);
        }
        pr[2 * q + e] = sr;
        pi[2 * q + e] = si;
      }
    }

    // Stage 2: D += A[:, 4s:4s+4] x B[4s:4s+4, :], K=32 total, fp32 exact.
    // B layout mirrors A: lane holds col N=lane%16, v0/v1 = K=(4s+2*half)+{0,1}.
    v8f acc = {0.f, 0.f, 0.f, 0.f, 0.f, 0.f, 0.f, 0.f};
#pragma unroll
    for (int s = 0; s < 4; ++s) {             // kk = 4s..4s+3  -> P_re rows
      v2f b = (v2f){pr[2 * s], pr[2 * s + 1]};
      acc = __builtin_amdgcn_wmma_f32_16x16x4_f32(
          false, afrag[s], false, b, (short)0, acc, false, false);
    }
#pragma unroll
    for (int s = 4; s < 8; ++s) {             // kk = 16+...    -> P_im rows
      v2f b = (v2f){pi[2 * (s - 4)], pi[2 * (s - 4) + 1]};
      acc = __builtin_amdgcn_wmma_f32_16x16x4_f32(
          false, afrag[s], false, b, (short)0, acc, false, false);
    }

    // D[g + 8*half, N]: half=0 -> Re(out[i=g]), half=1 -> Im(out[i=g]).
    // Lanes 0-8 and 16-24 together cover the contiguous 72B of out[i,x,:,:].
    if (M < 9) {
#pragma unroll
      for (int g = 0; g < 8; ++g) {
        out[(((size_t)g * X_SITES + x) * 9 + M) * 2 + half] = acc[g];
      }
    }
  }
}

extern "C" void kernel_launch(void* const* d_in, const int* in_sizes, int n_in,
                              void* d_out, int out_size, void* d_ws, size_t ws_size,
                              hipStream_t stream) {
  const float* f1 = (const float*)d_in[0];   // features_in1, complex64 flat
  const float* f2 = (const float*)d_in[1];   // features_in2, complex64 flat
  const float* w  = (const float*)d_in[2];   // weights,      complex64 flat
  float* out = (float*)d_out;

  // 4096 blocks x 8 waves = 32768 waves; grid-stride, 4 sites per wave.
  dim3 grid(4096), block(256);
  hipLaunchKernelGGL(lge_bilinear_wmma_kernel, grid, block, 0, stream,
                     f1, f2, w, out);
}